// LinearDecayMixup_57251914056262
// MI455X (gfx1250) — compile-verified
//
#include <hip/hip_runtime.h>
#include <hip/hip_bf16.h>
#include <stdint.h>

// ---------------------------------------------------------------------------
// LinearDecayMixup for MI455X (gfx1250).
//
// Pure streaming op: out[row] = mask(row) ? one_hot(label[row]) : pred[row].
// HBM roofline: 192 MiB written + ~96 MiB read (masked rows skip the read)
// => ~12.4 us @ 23.3 TB/s. No matmul content -> no WMMA; the CDNA5-specific
// path used is the async global<->LDS mover (ASYNCcnt) with NT cache hints.
//
// Layout: 1 block per token row (B*N = 65536 blocks), 192 threads = C/4,
// one B128 (float4) per thread.
// ---------------------------------------------------------------------------

typedef float v4f __attribute__((ext_vector_type(4)));

#define STAGE1_RATE 0.2
#define STAGE2_RATE 0.6

__global__ void __launch_bounds__(256)
linear_decay_mixup_kernel(const float* __restrict__ pred,    // [rows, C] f32
                          const int*   __restrict__ labels,  // [rows]
                          const float* __restrict__ rnd,     // [rows]
                          const int*   __restrict__ cur_p,   // [1]
                          const int*   __restrict__ tot_p,   // [1]
                          float*       __restrict__ out,     // [rows, C]
                          int C)
{
    extern __shared__ char lds_raw[];   // blockDim.x * 16 bytes

    const int row = blockIdx.x;
    const int tid = threadIdx.x;

    // --- schedule ratio (uniform; python-scalar double math) ----------------
    const double cur = (double)cur_p[0];
    const double tot = (double)tot_p[0];
    float ratio;
    if (cur < tot * STAGE1_RATE)      ratio = 1.0f;
    else if (cur < tot * STAGE2_RATE) ratio = (float)((tot * STAGE2_RATE - cur) /
                                                      ((STAGE2_RATE - STAGE1_RATE) * tot));
    else                              ratio = 0.0f;

    // --- per-row mask (block-uniform: no divergence inside the block) -------
    const int   label = labels[row];
    const float r     = rnd[row];
    const bool  mask  = (r < ratio) && (label >= 0);

    const size_t rowOff = (size_t)row * (size_t)C;
    v4f* __restrict__ dst4 = (v4f*)(out + rowOff) + tid;

    if (mask) {
        // One-hot row: never touch pred (saves ~50% of HBM read traffic).
        int cl = label;
        if (cl < 0)      cl = 0;
        if (cl > C - 1)  cl = C - 1;
        v4f v = (v4f){0.0f, 0.0f, 0.0f, 0.0f};
        const int base = tid * 4;
        if (cl >= base && cl < base + 4) v[cl - base] = 1.0f;
        __builtin_nontemporal_store(v, dst4);               // global_store_b128 th:NT
    } else {
        // Streaming row copy via the CDNA5 Tensor/async data path:
        //   global -> LDS (async, NT)  ...wait ASYNCcnt...  LDS -> global (async, NT)
        // Each lane owns a private 16B LDS slot, so waiting on this wave's own
        // ASYNCcnt fully orders the LDS write before the LDS read; no barrier.
        // Low 32 bits of a generic shared pointer are the LDS byte address.
        const uint32_t lds_off = (uint32_t)(uintptr_t)(lds_raw + tid * 16);
        const uint64_t gsrc    = (uint64_t)(uintptr_t)((const v4f*)(pred + rowOff) + tid);
        const uint64_t gdst    = (uint64_t)(uintptr_t)dst4;

        asm volatile("global_load_async_to_lds_b128 %0, %1, off th:TH_LOAD_NT"
                     :: "v"(lds_off), "v"(gsrc) : "memory");
        asm volatile("s_wait_asynccnt 0" ::: "memory");
        asm volatile("global_store_async_from_lds_b128 %0, %1, off th:TH_STORE_NT"
                     :: "v"(gdst), "v"(lds_off) : "memory");
        asm volatile("s_wait_asynccnt 0" ::: "memory");
        // S_ENDPGM performs an implicit wait-idle as well.
    }
}

extern "C" void kernel_launch(void* const* d_in, const int* in_sizes, int n_in,
                              void* d_out, int out_size, void* d_ws, size_t ws_size,
                              hipStream_t stream) {
    // setup_inputs() order:
    //  0: obj_sem_cls_pred [B,N,C] f32
    //  1: obj_labels       [B,N]   int
    //  2: random_numer     [B,N]   f32
    //  3: cur_step         [1]     int
    //  4: total_steps      [1]     int
    const float* pred   = (const float*)d_in[0];
    const int*   labels = (const int*)  d_in[1];
    const float* rnd    = (const float*)d_in[2];
    const int*   curp   = (const int*)  d_in[3];
    const int*   totp   = (const int*)  d_in[4];
    float*       out    = (float*)d_out;

    const int rows = in_sizes[1];              // B*N = 65536
    const int C    = in_sizes[0] / rows;       // 768

    const int threads = C / 4;                 // 192 lanes = 6 wave32s, one B128 each
    const size_t shmem = (size_t)threads * 16; // one 16B LDS slot per lane

    linear_decay_mixup_kernel<<<rows, threads, shmem, stream>>>(
        pred, labels, rnd, curp, totp, out, C);
}